// Top2Gate_61538291417527
// MI455X (gfx1250) — compile-verified
//
#include <hip/hip_runtime.h>
#include <hip/hip_bf16.h>

typedef __attribute__((ext_vector_type(2))) float v2f;
typedef __attribute__((ext_vector_type(4))) float v4f;
typedef __attribute__((ext_vector_type(8))) float v8f;
typedef __attribute__((ext_vector_type(4))) int   v4i;

typedef __attribute__((address_space(1))) v4i v4i_g;   // global
typedef __attribute__((address_space(3))) v4i v4i_l;   // LDS

#define KC 128          // K-chunk staged in LDS
#define ROWS 32         // tokens per block
#define PAD 4           // LDS row pad: stride 132 == 4 (mod 64) -> conflict-free ds_load_b64

#if __has_builtin(__builtin_amdgcn_global_load_async_to_lds_b128)
#define USE_ASYNC_LDS 1
#else
#define USE_ASYNC_LDS 0
#endif

template <int N>
__device__ __forceinline__ void wait_asynccnt_le() {
#if USE_ASYNC_LDS
#if __has_builtin(__builtin_amdgcn_s_wait_asynccnt)
    __builtin_amdgcn_s_wait_asynccnt((unsigned short)N);
#else
    if constexpr (N == 0) asm volatile("s_wait_asynccnt 0x0" ::: "memory");
    else                  asm volatile("s_wait_asynccnt 0x4" ::: "memory");
#endif
#endif
}

// ---------------------------------------------------------------------------
// Kernel 1: router GEMM (f32 WMMA 16x16x4) + fused top-2 / softmax gating.
// Block = 256 threads (8 wave32s) owns 32 tokens x all 128 experts.
// Wave w owns experts [16w,16w+16) and TWO token tiles sharing each B fragment.
// x strip is staged into LDS with double-buffered async-to-LDS copies.
// ---------------------------------------------------------------------------
__global__ void __launch_bounds__(256)
gemm_gate_kernel(const float* __restrict__ x,    // T x D
                 const float* __restrict__ wg,   // E x D
                 float* __restrict__ gates,      // ws: T x E softmax probs
                 int*   __restrict__ ws_idx1,    // ws: T
                 int*   __restrict__ ws_idx2,    // ws: T
                 float* __restrict__ out_idx1,   // T
                 float* __restrict__ out_idx2,   // T
                 float* __restrict__ out_g1,     // T
                 float* __restrict__ out_g2,     // T
                 int T, int D, int E)
{
    const int row0 = blockIdx.x * ROWS;
    const int tid  = threadIdx.x;
    const int lane = tid & 31;
    const int wave = tid >> 5;          // 0..7
    const int e0   = wave * 16;

    // f32 WMMA A(16x4)/B(4x16) fragment layout (ISA 7.12.2):
    //   lanes 0-15 : M/N = lane,    K elems {0,1}
    //   lanes 16-31: M/N = lane-16, K elems {2,3}
    const int m    = lane & 15;
    const int koff = (lane >> 4) * 2;

    __shared__ float As[2][ROWS][KC + PAD];   // double-buffered x strip

    // --- per-thread staging descriptors: 4 x b128 lane-transfers per chunk ---
    const float* gsrc[4];
    int lrow[4], lcol[4];
#pragma unroll
    for (int u = 0; u < 4; ++u) {
        const int j = tid + u * 256;      // 0..1023
        lrow[u] = j >> 5;                 // 0..31
        lcol[u] = (j & 31) * 4;           // 0..124
        gsrc[u] = x + (size_t)(row0 + lrow[u]) * D + lcol[u];
    }

    auto issue_chunk = [&](int buf) {
#pragma unroll
        for (int u = 0; u < 4; ++u) {
#if USE_ASYNC_LDS
            __builtin_amdgcn_global_load_async_to_lds_b128(
                (v4i_g*)(void*)gsrc[u],
                (v4i_l*)(void*)&As[buf][lrow[u]][lcol[u]],
                0, 0);
#else
            *(v4f*)&As[buf][lrow[u]][lcol[u]] = *(const v4f*)gsrc[u];
#endif
            gsrc[u] += KC;
        }
    };

    const float* bp = wg + (size_t)(e0 + m) * D + koff;   // B walks wg row (L2-resident)

    v8f acc0 = {};   // token tile 0 (rows row0..row0+15)
    v8f acc1 = {};   // token tile 1 (rows row0+16..row0+31)

    const int nchunk = D / KC;            // 32
    int buf = 0;
    issue_chunk(0);

    for (int c = 0; c < nchunk; ++c) {
        if (c + 1 < nchunk) {
            issue_chunk(buf ^ 1);
            wait_asynccnt_le<4>();        // chunk c complete (next 4 still in flight)
        } else {
            wait_asynccnt_le<0>();
        }
        __syncthreads();

        const float* bpp = bp;
#pragma unroll 4
        for (int s = 0; s < KC / 4; ++s) {
            v2f b  = *(const v2f*)bpp; bpp += 4;
            v2f a0 = *(const v2f*)&As[buf][m][s * 4 + koff];
            v2f a1 = *(const v2f*)&As[buf][16 + m][s * 4 + koff];
            acc0 = __builtin_amdgcn_wmma_f32_16x16x4_f32(
                false, a0, false, b, (short)0, acc0, false, false);
            acc1 = __builtin_amdgcn_wmma_f32_16x16x4_f32(
                false, a1, false, b, (short)0, acc1, false, false);
        }
        bp = bpp;

        __syncthreads();                  // all waves done reading As[buf]
        buf ^= 1;
    }

    // C/D layout: VGPR r -> (M = r + (lane>>4)*8, N = lane&15)
    __shared__ float tile[ROWS][129];
    {
        const int N     = lane & 15;
        const int Mbase = (lane >> 4) * 8;
#pragma unroll
        for (int r = 0; r < 8; ++r) {
            tile[Mbase + r][e0 + N]      = acc0[r];
            tile[16 + Mbase + r][e0 + N] = acc1[r];
        }
    }
    __syncthreads();

    __shared__ float s_rm[ROWS];
    __shared__ float s_rinv[ROWS];
    if (tid < ROWS) {
        const int row = row0 + tid;
        float m1 = -3.4e38f; int i1 = 0;
        for (int e = 0; e < E; ++e) {
            float v = tile[tid][e];
            if (v > m1) { m1 = v; i1 = e; }   // strict '>': lowest index wins ties (top_k)
        }
        float m2 = -3.4e38f; int i2 = 0;
        for (int e = 0; e < E; ++e) {
            if (e == i1) continue;
            float v = tile[tid][e];
            if (v > m2) { m2 = v; i2 = e; }
        }
        float Z = 0.0f;
        for (int e = 0; e < E; ++e) Z += __expf(tile[tid][e] - m1);
        const float rinv = 1.0f / Z;
        s_rm[tid]   = m1;
        s_rinv[tid] = rinv;
        ws_idx1[row]  = i1;
        ws_idx2[row]  = i2;
        out_idx1[row] = (float)i1;
        out_idx2[row] = (float)i2;
        out_g1[row]   = rinv;                   // exp(m1-m1)/Z
        out_g2[row]   = __expf(m2 - m1) * rinv;
    }
    __syncthreads();

    // full softmax matrix to scratch (for me[e] column sums)
    for (int i = tid; i < ROWS * E; i += 256) {
        const int r = i / E;
        const int e = i - r * E;
        gates[(size_t)(row0 + r) * E + e] = __expf(tile[r][e] - s_rm[r]) * s_rinv[r];
    }
}

// ---------------------------------------------------------------------------
// Kernel 2: per-expert token-order prefix scans (locations1/2 with capacity),
// fused me[e] column reduction. One block (128 thr = 4 wave32) per expert.
// ---------------------------------------------------------------------------
__global__ void __launch_bounds__(128)
scan_kernel(const int*   __restrict__ idx1,
            const int*   __restrict__ idx2,
            const float* __restrict__ gates,    // T x E
            float* __restrict__ out_loc1,       // T
            float* __restrict__ out_loc2,       // T
            float* __restrict__ laux_part,      // E  (me[e]*ce[e])
            int T, int E, int capacity)
{
    const int e    = blockIdx.x;
    const int tid  = threadIdx.x;     // 0..127
    const int lane = tid & 31;
    const int wave = tid >> 5;        // 0..3

    __shared__ int   wcnt[4];
    __shared__ float red[128];

    float me_local = 0.0f;

    // ---- pass 1: scan mask1, write locations1_s, accumulate me ----
    int base = 0;
    for (int t0 = 0; t0 < T; t0 += 128) {
        const int tok  = t0 + tid;
        const int flag = (idx1[tok] == e) ? 1 : 0;
        me_local += gates[(size_t)tok * E + e];

        const unsigned bal = (unsigned)__ballot(flag);   // wave32: low 32 bits
        const int lanepre  = __popc(bal & ((1u << lane) - 1u));
        if (lane == 0) wcnt[wave] = __popc(bal);
        __syncthreads();

        int waveoff = 0, chunktot = 0;
        for (int w = 0; w < 4; ++w) {
            if (w < wave) waveoff += wcnt[w];
            chunktot += wcnt[w];
        }
        if (flag) {
            const int pos = base + waveoff + lanepre;
            out_loc1[tok] = (pos < capacity) ? (float)pos : 0.0f;
        }
        base += chunktot;
        __syncthreads();
    }
    const int total1 = base;          // sum(mask1) over all tokens for this expert

    // ---- pass 2: scan mask2, offset by total1, write locations2_s ----
    base = 0;
    for (int t0 = 0; t0 < T; t0 += 128) {
        const int tok  = t0 + tid;
        const int flag = (idx2[tok] == e) ? 1 : 0;

        const unsigned bal = (unsigned)__ballot(flag);
        const int lanepre  = __popc(bal & ((1u << lane) - 1u));
        if (lane == 0) wcnt[wave] = __popc(bal);
        __syncthreads();

        int waveoff = 0, chunktot = 0;
        for (int w = 0; w < 4; ++w) {
            if (w < wave) waveoff += wcnt[w];
            chunktot += wcnt[w];
        }
        if (flag) {
            const int pos = base + waveoff + lanepre + total1;
            out_loc2[tok] = (pos < capacity) ? (float)pos : 0.0f;
        }
        base += chunktot;
        __syncthreads();
    }

    // ---- reduce me over block, emit me*ce ----
    red[tid] = me_local;
    __syncthreads();
    for (int s = 64; s > 0; s >>= 1) {
        if (tid < s) red[tid] += red[tid + s];
        __syncthreads();
    }
    if (tid == 0) {
        const float ce = (float)((total1 < capacity) ? total1 : capacity);
        laux_part[e] = red[0] * ce;
    }
}

// ---------------------------------------------------------------------------
// Kernel 3: l_aux = sum(me*ce) * E / T^2, plus scalar outputs.
// ---------------------------------------------------------------------------
__global__ void __launch_bounds__(128)
finalize_kernel(const float* __restrict__ laux_part,
                float* __restrict__ out, int T, int E, int capacity)
{
    __shared__ float red[128];
    const int tid = threadIdx.x;
    red[tid] = (tid < E) ? laux_part[tid] : 0.0f;
    __syncthreads();
    for (int s = 64; s > 0; s >>= 1) {
        if (tid < s) red[tid] += red[tid + s];
        __syncthreads();
    }
    if (tid == 0) {
        out[0] = red[0] * (float)E / ((float)T * (float)T);
        out[1] = (float)capacity;
        out[2] = (float)E;
    }
}

// ---------------------------------------------------------------------------
extern "C" void kernel_launch(void* const* d_in, const int* in_sizes, int n_in,
                              void* d_out, int out_size, void* d_ws, size_t ws_size,
                              hipStream_t stream)
{
    (void)n_in; (void)out_size; (void)ws_size;

    const float* x  = (const float*)d_in[0];
    const float* wg = (const float*)d_in[1];

    const int D = 4096;
    const int E = 128;
    const int T = in_sizes[0] / D;                 // 16384
    const int capacity = 2 * ((T + E - 1) / E);    // 256

    float* out = (float*)d_out;
    float* out_idx1 = out + 3;
    float* out_idx2 = out_idx1 + T;
    float* out_loc1 = out_idx2 + T;
    float* out_loc2 = out_loc1 + T;
    float* out_g1   = out_loc2 + T;
    float* out_g2   = out_g1 + T;

    float* gates     = (float*)d_ws;               // T*E floats (8 MB)
    float* laux_part = gates + (size_t)T * E;      // E floats
    int*   ws_idx1   = (int*)(laux_part + E);      // T ints
    int*   ws_idx2   = ws_idx1 + T;                // T ints

    gemm_gate_kernel<<<T / ROWS, 256, 0, stream>>>(
        x, wg, gates, ws_idx1, ws_idx2,
        out_idx1, out_idx2, out_g1, out_g2, T, D, E);

    scan_kernel<<<E, 128, 0, stream>>>(
        ws_idx1, ws_idx2, gates, out_loc1, out_loc2, laux_part, T, E, capacity);

    finalize_kernel<<<1, 128, 0, stream>>>(laux_part, out, T, E, capacity);
}